// SingleHeadedAttention_17136919511674
// MI455X (gfx1250) — compile-verified
//
#include <hip/hip_runtime.h>
#include <math.h>

// ---------------------------------------------------------------------------
// Shapes (fixed by the reference):
//   x: (8, 4096, 1024) f32; Wqk: (1024,128); Wv: (1024,1024)
//   gamma/beta: (2,128); rel_emb: (32,1)
//   CHUNK=512 -> 64 chunks of 512 rows; out: (8,4096,1024) f32
// Memory-bound problem (~470MB @ 23.3TB/s): full fp32 WMMA (16x16x4) is free,
// staging uses gfx1250 async global->LDS B128 copies (ASYNCcnt).
// ---------------------------------------------------------------------------

typedef __attribute__((ext_vector_type(2))) float v2f;
typedef __attribute__((ext_vector_type(4))) float v4f;
typedef __attribute__((ext_vector_type(8))) float v8f;

// D = A(16x4) * B(4x16) + C, full fp32 via V_WMMA_F32_16X16X4_F32.
static __device__ __forceinline__ v8f wmma4(v2f a, v2f b, v8f c) {
  return __builtin_amdgcn_wmma_f32_16x16x4_f32(
      /*neg_a=*/false, a, /*neg_b=*/false, b,
      /*c_mod=*/(short)0, c, /*reuse_a=*/false, /*reuse_b=*/false);
}

// Async copy of 16 bytes: global (SGPR base + per-lane byte offset) -> LDS.
// LDS byte address = low 32 bits of the flat address of the shared-mem slot.
static __device__ __forceinline__ void async_b128(const void* sbase,
                                                  unsigned goff_bytes,
                                                  void* lds_ptr) {
  unsigned ldsoff = (unsigned)(uintptr_t)lds_ptr;
  asm volatile("global_load_async_to_lds_b128 %0, %1, %2"
               :: "v"(ldsoff), "v"(goff_bytes), "s"(sbase) : "memory");
}

static __device__ __forceinline__ void async_wait_all() {
  asm volatile("s_wait_asynccnt 0x0" ::: "memory");
}

// ---------------------------------------------------------------------------
// Kernel 1: C = swish(X @ W + bias), optional q/k epilogue.
//   mode 0: out0[m][n] = swish(c)
//   mode 1: out0 = swish(c)*gamma[0]+beta[0]; out1 = swish(c)*gamma[1]+beta[1]
// Tile: 64(M) x 64(N), BK=32, 256 threads = 8 waves, 2 16x16 tiles per wave.
// ---------------------------------------------------------------------------
__global__ __launch_bounds__(256) void proj_kernel(
    const float* __restrict__ X, const float* __restrict__ W,
    const float* __restrict__ bias, const float* __restrict__ gamma,
    const float* __restrict__ beta, float* __restrict__ out0,
    float* __restrict__ out1, int K, int N, int mode)
{
  __shared__ float As[64][68];   // 17.4 KB, stride 68: A-style reads, bank step 4
  __shared__ float Bs[32][72];   // 9.2 KB, stride 72: B-style reads, halves +16 banks

  const int tid  = threadIdx.x;
  const int wave = tid >> 5, lane = tid & 31;
  const int half = lane >> 4, l16 = lane & 15;
  const int mt  = wave & 3;            // M sub-tile 0..3
  const int nt0 = (wave >> 2) * 2;     // two N sub-tiles per wave
  const int mb = blockIdx.y, nb = blockIdx.x;

  v8f acc[2] = {};

  for (int k0 = 0; k0 < K; k0 += 32) {
    const float* xbase = X + (size_t)(mb * 64) * K + k0;
#pragma unroll
    for (int e = tid; e < 64 * 8; e += 256) {      // 64 rows x 8 quads
      int r = e >> 3, c4 = e & 7;
      async_b128(xbase, (unsigned)((r * K + c4 * 4) * 4), &As[r][c4 * 4]);
    }
    const float* wbase = W + (size_t)k0 * N + nb * 64;
#pragma unroll
    for (int e = tid; e < 32 * 16; e += 256) {     // 32 rows x 16 quads
      int r = e >> 4, c4 = e & 15;
      async_b128(wbase, (unsigned)((r * N + c4 * 4) * 4), &Bs[r][c4 * 4]);
    }
    async_wait_all();
    __syncthreads();
#pragma unroll
    for (int kk = 0; kk < 32; kk += 4) {
      v2f a;
      a.x = As[mt * 16 + l16][kk + 2 * half + 0];
      a.y = As[mt * 16 + l16][kk + 2 * half + 1];
#pragma unroll
      for (int t = 0; t < 2; ++t) {
        v2f b;
        b.x = Bs[kk + 2 * half + 0][(nt0 + t) * 16 + l16];
        b.y = Bs[kk + 2 * half + 1][(nt0 + t) * 16 + l16];
        acc[t] = wmma4(a, b, acc[t]);
      }
    }
    __syncthreads();
  }

#pragma unroll
  for (int t = 0; t < 2; ++t) {
    int n = nb * 64 + (nt0 + t) * 16 + l16;
    float bn = bias[n];
#pragma unroll
    for (int r = 0; r < 8; ++r) {
      int m = mb * 64 + mt * 16 + 8 * half + r;
      float cv = acc[t][r] + bn;
      float s = cv / (1.0f + expf(-cv));   // swish
      if (mode == 0) {
        out0[(size_t)m * N + n] = s;
      } else {
        out0[(size_t)m * N + n] = s * gamma[n]     + beta[n];
        out1[(size_t)m * N + n] = s * gamma[N + n] + beta[N + n];
      }
    }
  }
}

// ---------------------------------------------------------------------------
// Kernel 2: sim = (q @ k^T) * 1024^-0.5 + rel_bias, causal-masked.
// Grid: (ktile=8, qtile=8, chunk=64). Each WG: 64x64 tile, K=128 staged by 64.
// K is staged ROW-MAJOR; the B fragment of q@k^T is read with A-style
// indexing (b = Ks[j][d]), so no transpose staging is needed.
// ---------------------------------------------------------------------------
__global__ __launch_bounds__(256) void sim_kernel(
    const float* __restrict__ Q, const float* __restrict__ Km,
    const float* __restrict__ rel, float* __restrict__ attn)
{
  const int ktile = blockIdx.x, qtile = blockIdx.y, z = blockIdx.z;
  const int bi = z >> 3, ck = z & 7;
  const int tid = threadIdx.x;

  if (ktile > qtile) {  // entire tile is masked out (uniform branch, no WMMA)
    const float ninf = -__builtin_inff();
    const v4f f = {ninf, ninf, ninf, ninf};
    for (int e = tid; e < 64 * 16; e += 256) {
      int i = qtile * 64 + (e >> 4);
      int j4 = e & 15;
      *(v4f*)&attn[((size_t)z * 512 + i) * 512 + ktile * 64 + j4 * 4] = f;
    }
    return;
  }

  __shared__ float Qs[64][68];   // 17.4 KB
  __shared__ float Ks[64][68];   // 17.4 KB, row-major [j][d]

  const int wave = tid >> 5, lane = tid & 31;
  const int half = lane >> 4, l16 = lane & 15;
  const int mt  = wave & 3;
  const int nt0 = (wave >> 2) * 2;
  const int rowbase = bi * 4096 + ck * 512;

  v8f acc[2] = {};

  for (int k0 = 0; k0 < 128; k0 += 64) {
    const float* qbase = Q + (size_t)(rowbase + qtile * 64) * 128 + k0;
    const float* kbase = Km + (size_t)(rowbase + ktile * 64) * 128 + k0;
#pragma unroll
    for (int e = tid; e < 64 * 16; e += 256) {     // 64 rows x 16 quads
      int r = e >> 4, c4 = e & 15;
      unsigned goff = (unsigned)((r * 128 + c4 * 4) * 4);
      async_b128(qbase, goff, &Qs[r][c4 * 4]);
      async_b128(kbase, goff, &Ks[r][c4 * 4]);
    }
    async_wait_all();
    __syncthreads();
#pragma unroll
    for (int kk = 0; kk < 64; kk += 4) {
      v2f a;
      a.x = Qs[mt * 16 + l16][kk + 2 * half + 0];
      a.y = Qs[mt * 16 + l16][kk + 2 * half + 1];
#pragma unroll
      for (int t = 0; t < 2; ++t) {
        v2f b;   // B[d][j] = K[j][d] -> A-style read of row-major Ks
        b.x = Ks[(nt0 + t) * 16 + l16][kk + 2 * half + 0];
        b.y = Ks[(nt0 + t) * 16 + l16][kk + 2 * half + 1];
        acc[t] = wmma4(a, b, acc[t]);
      }
    }
    __syncthreads();
  }

  const float scale = 0.03125f;                 // 1024^-0.5
  const float relscale = 11.313708498984761f;   // sqrt(128)
  const float inv_log8 = 16.0f / 2.0794415416798357f;  // (32-16)/log(128/16)

#pragma unroll
  for (int t = 0; t < 2; ++t) {
    int j = ktile * 64 + (nt0 + t) * 16 + l16;
#pragma unroll
    for (int r = 0; r < 8; ++r) {
      int i = qtile * 64 + mt * 16 + 8 * half + r;
      float v = acc[t][r] * scale;
      int nd = i - j;  // n = max(i - j, 0); masked when j > i
      if (nd < 0) {
        v = -__builtin_inff();
      } else {
        int bucket;
        if (nd < 16) {
          bucket = nd;
        } else {
          float vl = 16.0f + logf((float)nd * 0.0625f) * inv_log8;
          bucket = (int)vl;
          if (bucket > 31) bucket = 31;
        }
        v += rel[bucket] * relscale;
      }
      attn[((size_t)z * 512 + i) * 512 + j] = v;
    }
  }
}

// ---------------------------------------------------------------------------
// Kernel 3: row softmax over 512-wide rows, one wave32 per row, B128 I/O.
// ---------------------------------------------------------------------------
__global__ __launch_bounds__(256) void softmax_kernel(float* __restrict__ attn)
{
  const int row  = blockIdx.x * 8 + (threadIdx.x >> 5);
  const int lane = threadIdx.x & 31;
  v4f* p = (v4f*)(attn + (size_t)row * 512);

  v4f vals[4];
  float m = -__builtin_inff();
#pragma unroll
  for (int t = 0; t < 4; ++t) {
    vals[t] = p[lane + 32 * t];
#pragma unroll
    for (int c = 0; c < 4; ++c) m = fmaxf(m, vals[t][c]);
  }
#pragma unroll
  for (int off = 16; off > 0; off >>= 1) m = fmaxf(m, __shfl_xor(m, off, 32));

  float s = 0.0f;
#pragma unroll
  for (int t = 0; t < 4; ++t)
#pragma unroll
    for (int c = 0; c < 4; ++c) { vals[t][c] = expf(vals[t][c] - m); s += vals[t][c]; }
#pragma unroll
  for (int off = 16; off > 0; off >>= 1) s += __shfl_xor(s, off, 32);

  const float inv = 1.0f / s;
#pragma unroll
  for (int t = 0; t < 4; ++t) {
#pragma unroll
    for (int c = 0; c < 4; ++c) vals[t][c] *= inv;
    p[lane + 32 * t] = vals[t];
  }
}

// ---------------------------------------------------------------------------
// Kernel 4: out = attn @ v. Grid: (vtile=8, qtile=8, chunk=64).
// Each WG: 64(q) x 128(v) tile, K=512 staged by 64; 4 WMMA accs per wave.
// ---------------------------------------------------------------------------
__global__ __launch_bounds__(256) void av_kernel(
    const float* __restrict__ attn, const float* __restrict__ V,
    float* __restrict__ out)
{
  const int vtile = blockIdx.x, qtile = blockIdx.y, z = blockIdx.z;
  const int bi = z >> 3, ck = z & 7;
  const int tid = threadIdx.x;

  __shared__ float As[64][68];    // 17.4 KB (attn tile, A-style reads)
  __shared__ float Bs[64][136];   // 34.8 KB (v tile, B-style reads, halves +16 banks)

  const int wave = tid >> 5, lane = tid & 31;
  const int half = lane >> 4, l16 = lane & 15;
  const int mt  = wave & 3;
  const int nt0 = (wave >> 2) * 4;   // four N sub-tiles per wave
  const int rowbase = bi * 4096 + ck * 512;

  v8f acc[4] = {};

  for (int k0 = 0; k0 < 512; k0 += 64) {
    const float* abase = attn + ((size_t)z * 512 + qtile * 64) * 512 + k0;
#pragma unroll
    for (int e = tid; e < 64 * 16; e += 256) {     // 64 rows x 16 quads
      int r = e >> 4, c4 = e & 15;
      async_b128(abase, (unsigned)((r * 512 + c4 * 4) * 4), &As[r][c4 * 4]);
    }
    const float* vbase = V + (size_t)(rowbase + k0) * 1024 + vtile * 128;
#pragma unroll
    for (int e = tid; e < 64 * 32; e += 256) {     // 64 rows x 32 quads
      int r = e >> 5, c4 = e & 31;
      async_b128(vbase, (unsigned)((r * 1024 + c4 * 4) * 4), &Bs[r][c4 * 4]);
    }
    async_wait_all();
    __syncthreads();
#pragma unroll
    for (int kk = 0; kk < 64; kk += 4) {
      v2f a;
      a.x = As[mt * 16 + l16][kk + 2 * half + 0];
      a.y = As[mt * 16 + l16][kk + 2 * half + 1];
#pragma unroll
      for (int t = 0; t < 4; ++t) {
        v2f b;
        b.x = Bs[kk + 2 * half + 0][(nt0 + t) * 16 + l16];
        b.y = Bs[kk + 2 * half + 1][(nt0 + t) * 16 + l16];
        acc[t] = wmma4(a, b, acc[t]);
      }
    }
    __syncthreads();
  }

#pragma unroll
  for (int t = 0; t < 4; ++t) {
    int n = vtile * 128 + (nt0 + t) * 16 + l16;
#pragma unroll
    for (int r = 0; r < 8; ++r) {
      int m = qtile * 64 + mt * 16 + 8 * half + r;
      out[(size_t)(rowbase + m) * 1024 + n] = acc[t][r];
    }
  }
}

// ---------------------------------------------------------------------------
// Launch. Workspace layout (fp32): q(32768x128) | k(32768x128) |
// v(32768x1024) | attn(64x512x512)  -> 224 MiB total.
// ---------------------------------------------------------------------------
extern "C" void kernel_launch(void* const* d_in, const int* in_sizes, int n_in,
                              void* d_out, int out_size, void* d_ws, size_t ws_size,
                              hipStream_t stream)
{
  const float* x     = (const float*)d_in[0];
  const float* Wqk   = (const float*)d_in[1];
  const float* bqk   = (const float*)d_in[2];
  const float* Wv    = (const float*)d_in[3];
  const float* bv    = (const float*)d_in[4];
  const float* gamma = (const float*)d_in[5];
  const float* beta  = (const float*)d_in[6];
  const float* rel   = (const float*)d_in[7];
  float* out = (float*)d_out;

  float* qbuf = (float*)d_ws;
  float* kbuf = qbuf + (size_t)32768 * 128;
  float* vbuf = kbuf + (size_t)32768 * 128;
  float* attn = vbuf + (size_t)32768 * 1024;

  // qk projection -> q, k (N=128)
  proj_kernel<<<dim3(2, 512), 256, 0, stream>>>(x, Wqk, bqk, gamma, beta,
                                                qbuf, kbuf, 1024, 128, 1);
  // v projection (N=1024)
  proj_kernel<<<dim3(16, 512), 256, 0, stream>>>(x, Wv, bv, nullptr, nullptr,
                                                 vbuf, nullptr, 1024, 1024, 0);
  // sim + bias + causal mask
  sim_kernel<<<dim3(8, 8, 64), 256, 0, stream>>>(qbuf, kbuf, rel, attn);
  // row softmax (64*512 rows, 8 rows per block)
  softmax_kernel<<<dim3(64 * 512 / 8), 256, 0, stream>>>(attn);
  // attn @ v
  av_kernel<<<dim3(8, 8, 64), 256, 0, stream>>>(attn, vbuf, out);
}